// GHMCLoss_79087527788872
// MI455X (gfx1250) — compile-verified
//
#include <hip/hip_runtime.h>

#define BINS 10

typedef __attribute__((ext_vector_type(2))) float v2f;
typedef __attribute__((ext_vector_type(8))) float v8f;
typedef __attribute__((ext_vector_type(4))) float f32x4;
typedef __attribute__((ext_vector_type(4))) int   i32x4;

// ---------------------------------------------------------------------------
// Pass 1: stream all 23.6M elements once. Per block, accumulate:
//   channels 0..9  : count_b  = sum(valid            | bin==b)
//   channels 10..19: S_b      = sum(bce*valid        | bin==b)
//   channel 20     : sum(label_weights)
// into partials[block*32 + channel]. (channels 21..31 unused padding)
// ---------------------------------------------------------------------------
__global__ __launch_bounds__(256) void ghmc_partial(
    const int*   __restrict__ labels,
    const float* __restrict__ logits,
    const float* __restrict__ weights,
    float*       __restrict__ partials,
    int n4, int ntotal)
{
    __shared__ float hist[8][2 * BINS];   // [wave][2*bin + {0:count,1:S}]
    __shared__ float sWsum;

    const int tid  = threadIdx.x;
    const int wave = tid >> 5;

    if (tid < 8 * 2 * BINS) ((float*)hist)[tid] = 0.0f;
    if (tid == 0) sWsum = 0.0f;
    __syncthreads();

    float* h = hist[wave];
    float wsum = 0.0f;

    const int stride = gridDim.x * blockDim.x;
    for (int i = blockIdx.x * blockDim.x + tid; i < n4; i += stride) {
        f32x4 x4 = __builtin_nontemporal_load((const f32x4*)logits  + i);
        i32x4 l4 = __builtin_nontemporal_load((const i32x4*)labels  + i);
        f32x4 w4 = __builtin_nontemporal_load((const f32x4*)weights + i);
#pragma unroll
        for (int c = 0; c < 4; ++c) {
            float x = x4[c];
            float w = w4[c];
            int   l = l4[c];
            // p_norm = (l==1) ? 1-sigmoid(x) : sigmoid(x) = sigmoid((1-2l)*x)
            float en = __expf(l ? x : -x);                 // exp(-t)
            float p  = __builtin_amdgcn_rcpf(1.0f + en);   // sigmoid(t)
            int bin = (int)(p * 10.0f);
            bin = bin < 0 ? 0 : (bin > BINS - 1 ? BINS - 1 : bin);
            // bce = max(x,0) - x*l + log1p(exp(-|x|))
            float ea  = __expf(-fabsf(x));
            float bce = fmaxf(x, 0.0f) - (l ? x : 0.0f) + __logf(1.0f + ea);
            float valid = (w > 0.0f) ? 1.0f : 0.0f;
            wsum += w;
            atomicAdd(&h[2 * bin + 0], valid);
            atomicAdd(&h[2 * bin + 1], valid * bce);
        }
    }

    // scalar tail (N not multiple of 4): block 0 / thread 0 only
    if (blockIdx.x == 0 && tid == 0) {
        for (int i = n4 * 4; i < ntotal; ++i) {
            float x = logits[i];
            float w = weights[i];
            int   l = labels[i];
            float en = __expf(l ? x : -x);
            float p  = __builtin_amdgcn_rcpf(1.0f + en);
            int bin = (int)(p * 10.0f);
            bin = bin < 0 ? 0 : (bin > BINS - 1 ? BINS - 1 : bin);
            float ea  = __expf(-fabsf(x));
            float bce = fmaxf(x, 0.0f) - (l ? x : 0.0f) + __logf(1.0f + ea);
            float valid = (w > 0.0f) ? 1.0f : 0.0f;
            wsum += w;
            atomicAdd(&h[2 * bin + 0], valid);
            atomicAdd(&h[2 * bin + 1], valid * bce);
        }
    }

    atomicAdd(&sWsum, wsum);
    __syncthreads();

    if (tid < 2 * BINS) {
        float v = 0.0f;
#pragma unroll
        for (int wv = 0; wv < 8; ++wv) v += hist[wv][tid];
        int bin = tid >> 1, sel = tid & 1;
        partials[blockIdx.x * 32 + sel * BINS + bin] = v;
    }
    if (tid == 20) partials[blockIdx.x * 32 + 20] = sWsum;
}

// ---------------------------------------------------------------------------
// Pass 2: single wave. Reduce G x 32 partials with V_WMMA_F32_16X16X4_F32:
// A = all-ones 16x4  =>  D[i][j] = sum_K B[K][j]  (exact f32 column sums,
// invariant to the K-slot<->row assignment in the B register layout).
// Two accumulators cover channels 0..15 and 16..31. Lane 0 then applies the
// GHM per-bin weighting epilogue.
// ---------------------------------------------------------------------------
__global__ __launch_bounds__(32) void ghmc_finalize(
    const float* __restrict__ partials, float* __restrict__ out, int G)
{
    const int lane = threadIdx.x;
    const int c    = lane & 15;
    const int sel  = lane >> 4;

    v2f a; a[0] = 1.0f; a[1] = 1.0f;   // ones matrix (16x4, 2 VGPRs/lane)
    v8f accLo = {};
    v8f accHi = {};

    for (int base = 0; base < G; base += 4) {
        const float* p0 = partials + (size_t)(base + 2 * sel) * 32;
        v2f bLo, bHi;
        bLo[0] = p0[c];          // row base+2*sel,   channel c
        bLo[1] = p0[32 + c];     // row base+2*sel+1, channel c
        bHi[0] = p0[16 + c];     // row base+2*sel,   channel 16+c
        bHi[1] = p0[48 + c];     // row base+2*sel+1, channel 16+c
        accLo = __builtin_amdgcn_wmma_f32_16x16x4_f32(
            false, a, false, bLo, (short)0, accLo, false, false);
        accHi = __builtin_amdgcn_wmma_f32_16x16x4_f32(
            false, a, false, bHi, (short)0, accHi, false, false);
    }

    __shared__ float tot[32];
    if (lane < 16) {
        tot[lane]      = accLo[0];   // D row 0 -> per-column totals, ch 0..15
        tot[16 + lane] = accHi[0];   // ch 16..31
    }
    __syncthreads();

    if (lane == 0) {
        float total = tot[20];
        total = total > 1.0f ? total : 1.0f;            // max(sum(w), 1)
        float sumS = 0.0f, nbins = 0.0f;
#pragma unroll
        for (int b = 0; b < BINS; ++b) {
            float cnt = tot[b];
            if (cnt > 0.0f) {
                nbins += 1.0f;
                float pw = total / (cnt > 1.0f ? cnt : 1.0f);
                sumS += pw * tot[10 + b];
            }
        }
        float res = sumS;
        if (nbins > 0.0f) res = res / nbins;            // beta / max(n,1)
        out[0] = res / total;
    }
}

extern "C" void kernel_launch(void* const* d_in, const int* in_sizes, int n_in,
                              void* d_out, int out_size, void* d_ws, size_t ws_size,
                              hipStream_t stream) {
    const int*   labels  = (const int*)d_in[0];
    const float* logits  = (const float*)d_in[1];
    const float* weights = (const float*)d_in[2];
    float* out      = (float*)d_out;
    float* partials = (float*)d_ws;

    const int N  = in_sizes[0];
    const int n4 = N / 4;

    int G = 1024;                                   // blocks == partial rows
    size_t need = (size_t)G * 32 * sizeof(float);
    if (ws_size < need) {                           // deterministic: ws_size fixed
        G = (int)(ws_size / (32 * sizeof(float)));
        G &= ~3;                                    // multiple of 4 for WMMA K
        if (G < 4) G = 4;
    }

    ghmc_partial <<<G, 256, 0, stream>>>(labels, logits, weights, partials, n4, N);
    ghmc_finalize<<<1,  32, 0, stream>>>(partials, out, G);
}